// MatchingPursuit_3032246911270
// MI455X (gfx1250) — compile-verified
//
#include <hip/hip_runtime.h>
#include <hip/hip_bf16.h>

typedef __attribute__((ext_vector_type(2))) float v2f;
typedef __attribute__((ext_vector_type(4))) float v4f;
typedef __attribute__((ext_vector_type(8))) float v8f;

static constexpr int   BATCH = 512;     // batch
static constexpr int   NF    = 1024;    // in_features
static constexpr int   NA    = 4096;    // out_features (atoms)
static constexpr float LAMBD = 0.2f;
static constexpr int   ADMM_ITERS   = 50;
static constexpr int   NEWTON_ITERS = 14;

// 32-bit LDS offset of a __shared__ object (flat aperture low bits == DS offset)
__device__ __forceinline__ unsigned ldsaddr(const void* p)
{
    return (unsigned)(unsigned long long)(uintptr_t)p;
}

// ---------------------------------------------------------------------------
// NT GEMM with fp32 WMMA:  C[M,N] = epi( sum_k A[m,k] * B[n,k] )
// A: M x K row-major, B: N x K row-major.  M%128==0, N%128==0, K%32==0.
// mode 0: C = acc
// mode 1: C = acc + I            (Gram matrix G = W^T W + I)
// mode 2: C = 2*D - acc          (Newton-Schulz: Xn = 2X - X*(G*X))
//
// Tile fill uses CDNA5 async global->LDS (no VGPR round trip, ASYNCcnt),
// double-buffered so tile n+1 streams in while tile n is consumed by WMMA.
// ---------------------------------------------------------------------------
__global__ __launch_bounds__(256)
void gemm_nt_f32_wmma(const float* __restrict__ A, const float* __restrict__ B,
                      float* __restrict__ C, const float* __restrict__ D,
                      int M, int N, int K, int mode)
{
    __shared__ float As[2][128][36];   // stride 36: 16B rows, conflict-free frags
    __shared__ float Bs[2][128][36];

    const int bm    = blockIdx.y * 128;
    const int bn    = blockIdx.x * 128;
    const int t     = threadIdx.x;          // 0..255 (8 waves of 32)
    const int wave  = t >> 5;
    const int lane  = t & 31;
    const int wm    = (wave >> 2) * 64;     // wave M offset: 0 / 64
    const int wn    = (wave & 3) * 32;      // wave N offset: 0/32/64/96
    const int half  = lane >> 4;            // 0: lanes 0-15, 1: lanes 16-31
    const int row16 = lane & 15;

    // per-thread fill coordinates: 4 x b128 for A-tile, 4 for B-tile
    const int c0 = (t & 7) << 2;            // col 0,4,...,28
    const int r0 = t >> 3;                  // row 0..31 (+32*i)

    v8f acc[4][2];
    v8f zero = {0.f,0.f,0.f,0.f,0.f,0.f,0.f,0.f};
#pragma unroll
    for (int i = 0; i < 4; ++i)
#pragma unroll
        for (int j = 0; j < 2; ++j) acc[i][j] = zero;

    auto issue_tile = [&](int bk, int buf) {
#pragma unroll
        for (int i = 0; i < 4; ++i) {
            int r = r0 + 32 * i;
            unsigned           lA = ldsaddr(&As[buf][r][c0]);
            unsigned           lB = ldsaddr(&Bs[buf][r][c0]);
            unsigned long long gA =
                (unsigned long long)(A + (size_t)(bm + r) * K + bk + c0);
            unsigned long long gB =
                (unsigned long long)(B + (size_t)(bn + r) * K + bk + c0);
            asm volatile("global_load_async_to_lds_b128 %0, %1, off"
                         :: "v"(lA), "v"(gA) : "memory");
            asm volatile("global_load_async_to_lds_b128 %0, %1, off"
                         :: "v"(lB), "v"(gB) : "memory");
        }
    };

    issue_tile(0, 0);
    asm volatile("s_wait_asynccnt 0x0" ::: "memory");
    __syncthreads();

    int cur = 0;
    for (int bk = 0; bk < K; bk += 32) {
        const bool more = (bk + 32) < K;
        if (more) issue_tile(bk + 32, cur ^ 1);   // stream next tile during compute

        const float (*as)[36] = As[cur];
        const float (*bs)[36] = Bs[cur];
#pragma unroll
        for (int k0 = 0; k0 < 32; k0 += 4) {
            // A frag (16x4): lane<16 holds A[m,k0..k0+1]; lane>=16 holds A[m,k0+2..k0+3]
            v2f af[4], bf[2];
#pragma unroll
            for (int i = 0; i < 4; ++i)
                af[i] = *(const v2f*)&as[wm + i * 16 + row16][k0 + 2 * half];
#pragma unroll
            for (int j = 0; j < 2; ++j)
                bf[j] = *(const v2f*)&bs[wn + j * 16 + row16][k0 + 2 * half];
#pragma unroll
            for (int i = 0; i < 4; ++i)
#pragma unroll
                for (int j = 0; j < 2; ++j)
                    acc[i][j] = __builtin_amdgcn_wmma_f32_16x16x4_f32(
                        false, af[i], false, bf[j], (short)0, acc[i][j],
                        false, false);
        }

        if (more) asm volatile("s_wait_asynccnt 0x0" ::: "memory");
        __syncthreads();
        cur ^= 1;
    }

    // Epilogue: v8f element e => M = e + 8*half, N = row16 (per 16x16 subtile)
#pragma unroll
    for (int i = 0; i < 4; ++i)
#pragma unroll
        for (int j = 0; j < 2; ++j) {
            int gm0 = bm + wm + i * 16 + 8 * half;
            int gn  = bn + wn + j * 16 + row16;
#pragma unroll
            for (int e = 0; e < 8; ++e) {
                int    gm  = gm0 + e;
                size_t off = (size_t)gm * N + gn;
                float  val = acc[i][j][e];
                if (mode == 1)      C[off] = val + ((gm == gn) ? 1.0f : 0.0f);
                else if (mode == 2) C[off] = 2.0f * D[off] - val;
                else                C[off] = val;
            }
        }
}

// ---------------------------------------------------------------------------
// Small helper kernels
// ---------------------------------------------------------------------------
__global__ __launch_bounds__(256)
void normalize_rows_k(const float* __restrict__ W, float* __restrict__ Wu)
{
    __shared__ float red[256];
    int row = blockIdx.x;
    const float* src = W + (size_t)row * NF;
    float s = 0.f;
    for (int c = threadIdx.x; c < NF; c += 256) { float v = src[c]; s += v * v; }
    red[threadIdx.x] = s; __syncthreads();
    for (int st = 128; st > 0; st >>= 1) {
        if (threadIdx.x < st) red[threadIdx.x] += red[threadIdx.x + st];
        __syncthreads();
    }
    float inv = 1.0f / sqrtf(red[0]);
    float* dst = Wu + (size_t)row * NF;
    for (int c = threadIdx.x; c < NF; c += 256) dst[c] = src[c] * inv;
}

__global__ __launch_bounds__(256)
void transpose_k(const float* __restrict__ src, float* __restrict__ dst,
                 int rows, int cols)   // dst[cols][rows] = src[rows][cols]^T
{
    __shared__ float tile[32][33];
    int bx = blockIdx.x * 32, by = blockIdx.y * 32;
    int tx = threadIdx.x, ty = threadIdx.y;   // 32 x 8
    for (int i = 0; i < 32; i += 8)
        tile[ty + i][tx] = src[(size_t)(by + ty + i) * cols + bx + tx];
    __syncthreads();
    for (int i = 0; i < 32; i += 8)
        dst[(size_t)(bx + ty + i) * rows + by + tx] = tile[tx][ty + i];
}

__global__ __launch_bounds__(256)
void zero_k(float* p, int n)
{
    int i = blockIdx.x * 256 + threadIdx.x;
    if (i < n) p[i] = 0.f;
}

__global__ __launch_bounds__(256)
void copy_k(const float* __restrict__ s, float* __restrict__ d, int n)
{
    int i = blockIdx.x * 256 + threadIdx.x;
    if (i < n) d[i] = s[i];
}

// Gershgorin bound: s = max_i sum_j |G[i][j]|  (>= lambda_max of SPD G)
__global__ __launch_bounds__(256)
void rowabs_max_k(const float* __restrict__ G, float* __restrict__ s)
{
    __shared__ float red[256];
    int row = blockIdx.x;
    float a = 0.f;
    for (int c = threadIdx.x; c < NF; c += 256)
        a += fabsf(G[(size_t)row * NF + c]);
    red[threadIdx.x] = a; __syncthreads();
    for (int st = 128; st > 0; st >>= 1) {
        if (threadIdx.x < st) red[threadIdx.x] += red[threadIdx.x + st];
        __syncthreads();
    }
    if (threadIdx.x == 0)
        atomicMax((unsigned int*)s, __float_as_uint(red[0]));  // all positive
}

__global__ __launch_bounds__(256)
void init_X_k(float* __restrict__ X, const float* __restrict__ s)
{
    int i = blockIdx.x * 256 + threadIdx.x;   // over NF*NF
    int r = i >> 10, c = i & (NF - 1);
    X[i] = (r == c) ? (1.0f / s[0]) : 0.0f;
}

// Fused ADMM elementwise: x = r - S; v' = soft(x+u); u' = (x+u) - v'; r' = Atb+v'-u'
__global__ __launch_bounds__(256)
void admm_update_k(const float* __restrict__ Atb, const float* __restrict__ S,
                   float* __restrict__ v, float* __restrict__ u,
                   float* __restrict__ r)
{
    int i = blockIdx.x * 256 + threadIdx.x;   // over BATCH*NA
    float atb = Atb[i], vv = v[i], uu = u[i];
    float rr = atb + vv - uu;
    float x  = rr - S[i];
    float vt = x + uu;
    float av = fabsf(vt) - LAMBD;
    float vn = (av > 0.f) ? copysignf(av, vt) : 0.f;
    float un = vt - vn;
    v[i] = vn;
    u[i] = un;
    r[i] = atb + vn - un;
}

// ---------------------------------------------------------------------------
static inline void gemm(const float* A, const float* B, float* C, const float* D,
                        int M, int N, int K, int mode, hipStream_t s)
{
    dim3 grid(N / 128, M / 128);
    gemm_nt_f32_wmma<<<grid, 256, 0, s>>>(A, B, C, D, M, N, K, mode);
}

extern "C" void kernel_launch(void* const* d_in, const int* in_sizes, int n_in,
                              void* d_out, int out_size, void* d_ws, size_t ws_size,
                              hipStream_t stream)
{
    const float* x_in = (const float*)d_in[0];   // (512,1,32,32) -> (512,1024)
    const float* w_in = (const float*)d_in[1];   // (4096,1024)
    float* out = (float*)d_out;                  // z (512*4096) ++ decoded (512*1024)

    // workspace partition (~92 MB, fits in the 192 MB L2 as a resident set)
    float* p   = (float*)d_ws;
    float* W   = p; p += (size_t)NA * NF;        // w_unit
    float* WT  = p; p += (size_t)NF * NA;        // w_unit^T
    float* G   = p; p += (size_t)NF * NF;        // I + W^T W
    float* X0  = p; p += (size_t)NF * NF;        // Newton ping
    float* X1  = p; p += (size_t)NF * NF;        // Newton pong
    float* T   = p; p += (size_t)NF * NF;        // Newton temp G*X
    float* Atb = p; p += (size_t)BATCH * NA;
    float* v   = p; p += (size_t)BATCH * NA;
    float* u   = p; p += (size_t)BATCH * NA;
    float* r   = p; p += (size_t)BATCH * NA;
    float* S   = p; p += (size_t)BATCH * NA;
    float* P   = p; p += (size_t)BATCH * NF;
    float* Q   = p; p += (size_t)BATCH * NF;
    float* sb  = p; p += 1;                      // Gershgorin scalar

    // 1. normalize rows of weight
    normalize_rows_k<<<NA, 256, 0, stream>>>(w_in, W);
    // 2. WT = W^T
    transpose_k<<<dim3(NF / 32, NA / 32), dim3(32, 8), 0, stream>>>(W, WT, NA, NF);
    // 3. Atb[b,j] = sum_k x[b,k] W[j,k]
    gemm(x_in, W, Atb, nullptr, BATCH, NA, NF, 0, stream);
    // 4. G = I + W^T W   (NT on WT: G[i,j] = sum_k WT[i,k] WT[j,k])
    gemm(WT, WT, G, nullptr, NF, NF, NA, 1, stream);

    // 5. zero state + scalar
    zero_k<<<(BATCH * NA + 255) / 256, 256, 0, stream>>>(v, BATCH * NA);
    zero_k<<<(BATCH * NA + 255) / 256, 256, 0, stream>>>(u, BATCH * NA);
    zero_k<<<1, 256, 0, stream>>>(sb, 1);

    // 6. Newton-Schulz inverse of G (all iterates symmetric => NT GEMMs)
    rowabs_max_k<<<NF, 256, 0, stream>>>(G, sb);
    init_X_k<<<(NF * NF) / 256, 256, 0, stream>>>(X0, sb);
    float* Xa = X0; float* Xb = X1;
    for (int it = 0; it < NEWTON_ITERS; ++it) {
        gemm(G,  Xa, T,  nullptr, NF, NF, NF, 0, stream);  // T  = G * X
        gemm(Xa, T,  Xb, Xa,      NF, NF, NF, 2, stream);  // Xn = 2X - X*T
        float* tmp = Xa; Xa = Xb; Xb = tmp;
    }
    const float* Ginv = Xa;

    // 7. ADMM iterations (Woodbury: x = r - ((r W) Ginv) W^T)
    const float* rcur = Atb;   // first iter: v=u=0 => r = Atb
    for (int it = 0; it < ADMM_ITERS; ++it) {
        gemm(rcur, WT,   P, nullptr, BATCH, NF, NA, 0, stream);  // P = r W
        gemm(P,    Ginv, Q, nullptr, BATCH, NF, NF, 0, stream);  // Q = P Ginv
        gemm(Q,    W,    S, nullptr, BATCH, NA, NF, 0, stream);  // S = Q W^T
        admm_update_k<<<(BATCH * NA) / 256, 256, 0, stream>>>(Atb, S, v, u, r);
        rcur = r;
    }

    // 8. outputs: z = v, decoded = z @ w_unit
    copy_k<<<(BATCH * NA) / 256, 256, 0, stream>>>(v, out, BATCH * NA);
    gemm(v, WT, out + (size_t)BATCH * NA, nullptr, BATCH, NF, NA, 0, stream);

    (void)in_sizes; (void)n_in; (void)out_size; (void)ws_size;
}